// CurrentLIFNetwork_66357244723845
// MI455X (gfx1250) — compile-verified
//
#include <hip/hip_runtime.h>

// ---------------------------------------------------------------------------
// LIF spiking network, B=8, T=500, N=4096 on MI455X (gfx1250).
// Per step: elementwise LIF update -> spikes, then (B x N) @ (N x N) matmul
// with rows routed to I_exc / I_inh by presynaptic type (exc rows first).
// W (64 MB fp32) stays resident in the 192 MB L2; the matmul is L2-BW bound,
// so fp32 WMMA (V_WMMA_F32_16X16X4_F32) is exact and fast enough.
// Hot-loop design: compile-time N specialization (immediate-offset W loads),
// two branch-free K loops (exc then inh), zero-padded 16-row spike buffer,
// unroll-8 with one pointer bump per group, software prefetch of the W stream.
// ---------------------------------------------------------------------------

typedef float v2f __attribute__((ext_vector_type(2)));
typedef float v8f __attribute__((ext_vector_type(8)));

#define TAU_SYN_E 0.005f
#define TAU_SYN_I 0.01f
#define TAU_MEM_E 0.02f
#define TAU_MEM_I 0.02f
#define U_REST_E (-65.0f)
#define U_REST_I (-65.0f)
#define THETA_E (-50.0f)
#define THETA_I (-50.0f)
#define U_RESET_E (-65.0f)
#define U_RESET_I (-65.0f)
#define R_E 0.1f
#define R_I 0.1f

// ------------------------- state init (once per launch) --------------------
// Copies initial state into ws and counts excitatory neurons (exc region is
// contiguous at the front, so the count equals the split point n_exc).
__global__ void lif_init(const float* __restrict__ iv,
                         const float* __restrict__ iie,
                         const float* __restrict__ iii,
                         const int* __restrict__ types,
                         float* __restrict__ st_v,
                         float* __restrict__ st_ie,
                         float* __restrict__ st_ii,
                         int* __restrict__ nexc_cnt,
                         int BN, int N) {
  int i = blockIdx.x * blockDim.x + threadIdx.x;
  if (i < BN) {
    st_v[i]  = iv[i];
    st_ie[i] = iie[i];
    st_ii[i] = iii[i];
  }
  if (i < N && types[i] == 1) {
    atomicAdd(nexc_cnt, 1);
  }
}

// ------------------- elementwise LIF update + spike ------------------------
// Also applies the synaptic decay (I *= alpha) so the matmul kernel only has
// to accumulate its contribution. CN > 0 = compile-time N.
template <int CN>
__global__ void lif_update_spike(const float* __restrict__ delta_t,
                                 const int* __restrict__ types,
                                 float* __restrict__ st_v,
                                 float* __restrict__ st_ie,
                                 float* __restrict__ st_ii,
                                 float* __restrict__ spk,
                                 float* __restrict__ out_s,
                                 float* __restrict__ out_v,
                                 int N_rt, int T, int t, int BN) {
  const int N = (CN > 0) ? CN : N_rt;
  int i = blockIdx.x * blockDim.x + threadIdx.x;
  if (i >= BN) return;
  int b = i / N;
  int n = i - b * N;

  float dt      = delta_t[0] * 0.001f;  // ms -> s
  float alpha_E = __expf(-dt / TAU_SYN_E);
  float alpha_I = __expf(-dt / TAU_SYN_I);

  bool  exc    = (types[n] == 1);
  float beta   = exc ? __expf(-dt / TAU_MEM_E) : __expf(-dt / TAU_MEM_I);
  float urest  = exc ? U_REST_E : U_REST_I;
  float theta  = exc ? THETA_E : THETA_I;
  float ureset = exc ? U_RESET_E : U_RESET_I;
  float drive  = (exc ? R_E : R_I) * (1.0f - beta);

  float v = st_v[i];
  float I = st_ie[i] + st_ii[i];
  v = urest + (v - urest) * beta + I * drive;
  float s = (v >= theta) ? 1.0f : 0.0f;
  v = (s > 0.0f) ? ureset : v;

  st_v[i] = v;
  spk[i]  = s;   // rows B..15 of spk were zeroed once per launch

  size_t o = (size_t)b * T * N + (size_t)t * N + n;
  out_s[o] = s;
  out_v[o] = v;

  st_ie[i] *= alpha_E;  // pre-decay; matmul kernel adds contribution
  st_ii[i] *= alpha_I;
}

// ---------------------------- WMMA matmul ----------------------------------
// One wave per 16-column tile. A = spikes (16x4 fp32, rows 8..15 are a
// zero-filled pad in the spike buffer), B = W rows (4x16 fp32).
// Two branch-free K loops: [0, nexc) -> acc_e, [nexc, N) -> acc_i.
// CN > 0 makes all W strides compile-time -> immediate-offset loads.
template <int CN>
__global__ __launch_bounds__(128)
void lif_matmul(const float* __restrict__ W,
                const int* __restrict__ types,
                const float* __restrict__ Ew,
                const float* __restrict__ Iw,
                const float* __restrict__ spk,
                const int* __restrict__ nexc_cnt,
                float* __restrict__ st_ie,
                float* __restrict__ st_ii,
                float* __restrict__ out_ie,
                float* __restrict__ out_ii,
                int N_rt, int T, int t, int B) {
  const int N = (CN > 0) ? CN : N_rt;

  int wid  = threadIdx.x >> 5;   // wave within block (wave32)
  int lane = threadIdx.x & 31;
  int half = lane >> 4;          // 0: K pair {0,1}, 1: K pair {2,3}
  int m    = lane & 15;          // M for A / column within tile for B

  int tile = blockIdx.x * 4 + wid;
  int col  = tile * 16 + m;

  int nexc        = nexc_cnt[0];       // exc/inh split (multiple of 4)
  int nexc_chunks = nexc >> 2;
  int tot_chunks  = N >> 2;

  // A fragment source: this lane's spike row, K pair selected by half-wave.
  const float* sp = spk + (size_t)m * N + 2 * half;
  // B fragment source: W row (2*half), column col; wp[N] is row (2*half)+1.
  // With CN compile-time, wp[u*4N] and wp[u*4N + N] are immediate offsets.
  const float* wp = W + (size_t)(2 * half) * N + col;

  const size_t wstep  = (size_t)4 * N;    // K-chunk advance in W
  const size_t pfdist = 8 * wstep;        // prefetch ~8 chunks (32 rows) ahead

  v8f acc_e = {};
  v8f acc_i = {};

#pragma unroll 8
  for (int c = 0; c < nexc_chunks; ++c) {
    v2f a = *(const v2f*)sp;              // global_load_b64 (8B aligned)
    v2f bf;
    bf.x = wp[0];
    bf.y = wp[(size_t)N];
    __builtin_prefetch(wp + pfdist, 0, 1);  // W streams: no L0 reuse
    acc_e = __builtin_amdgcn_wmma_f32_16x16x4_f32(
        false, a, false, bf, (short)0, acc_e, false, false);
    sp += 4;
    wp += wstep;
  }

#pragma unroll 8
  for (int c = nexc_chunks; c < tot_chunks; ++c) {
    v2f a = *(const v2f*)sp;
    v2f bf;
    bf.x = wp[0];
    bf.y = wp[(size_t)N];
    __builtin_prefetch(wp + pfdist, 0, 1);
    acc_i = __builtin_amdgcn_wmma_f32_16x16x4_f32(
        false, a, false, bf, (short)0, acc_i, false, false);
    sp += 4;
    wp += wstep;
  }

  // Postsynaptic column scale applied once at the end.
  float scale = (types[col] == 1) ? Ew[0] : Iw[0];

  // C/D layout: VGPR r, lanes 0-15 -> M=r, lanes 16-31 -> M=r+8.
  // Real batches live in M=0..7 -> lanes 0-15 only.
  if (half == 0) {
    for (int r = 0; r < 8; ++r) {
      if (r < B) {
        int    si = r * N + col;
        float  ie = st_ie[si] + acc_e[r] * scale;
        float  ii = st_ii[si] + acc_i[r] * scale;
        st_ie[si] = ie;
        st_ii[si] = ii;
        size_t o  = (size_t)r * T * N + (size_t)t * N + col;
        out_ie[o] = ie;
        out_ii[o] = ii;
      }
    }
  }
}

// ---------------------------------------------------------------------------
extern "C" void kernel_launch(void* const* d_in, const int* in_sizes, int n_in,
                              void* d_out, int out_size, void* d_ws, size_t ws_size,
                              hipStream_t stream) {
  // Inputs (setup_inputs order):
  // 0: n_steps (int, device - unused; T derived from sizes)
  // 1: delta_t (f32 scalar)      2: neuron_types (int32, N)
  // 3: recurrent_weights (f32 NxN)
  // 4: E_weight (f32)            5: I_weight (f32)
  // 6: initial_v (f32 BxN)       7: initial_I_exc    8: initial_I_inh
  const float* delta_t = (const float*)d_in[1];
  const int*   types   = (const int*)d_in[2];
  const float* W       = (const float*)d_in[3];
  const float* Ew      = (const float*)d_in[4];
  const float* Iw      = (const float*)d_in[5];
  const float* iv      = (const float*)d_in[6];
  const float* iie     = (const float*)d_in[7];
  const float* iii     = (const float*)d_in[8];

  int N  = in_sizes[2];
  int BN = in_sizes[6];
  int B  = BN / N;
  long long BTN = (long long)out_size / 4;   // elements per output tensor
  int T  = (int)(BTN / BN);

  float* out    = (float*)d_out;
  float* out_s  = out;
  float* out_v  = out + BTN;
  float* out_ie = out + 2 * BTN;
  float* out_ii = out + 3 * BTN;

  // Workspace: v, I_exc, I_inh states (B x N each), 16-row spike buffer
  // (rows B..15 zero padding for the WMMA M dimension), exc counter.
  float* ws    = (float*)d_ws;
  float* st_v  = ws;
  float* st_ie = ws + BN;
  float* st_ii = ws + 2 * BN;
  float* spk   = ws + 3 * BN;                 // 16 * N floats
  int*   ncnt  = (int*)(ws + 3 * BN + 16 * (size_t)N);

  // Zero the exc counter and the spike-pad rows (graph-capturable memsets).
  hipMemsetAsync(ncnt, 0, sizeof(int), stream);
  hipMemsetAsync(spk + (size_t)B * N, 0, (size_t)(16 - B) * N * sizeof(float), stream);

  lif_init<<<(BN + 255) / 256, 256, 0, stream>>>(
      iv, iie, iii, types, st_v, st_ie, st_ii, ncnt, BN, N);

  int tiles   = N / 16;       // 256 column tiles
  int blocks2 = tiles / 4;    // 4 waves (tiles) per 128-thread block
  int blocks1 = (BN + 255) / 256;

  const bool fast = (N == 4096);
  for (int t = 0; t < T; ++t) {
    if (fast) {
      lif_update_spike<4096><<<blocks1, 256, 0, stream>>>(
          delta_t, types, st_v, st_ie, st_ii, spk, out_s, out_v, N, T, t, BN);
      lif_matmul<4096><<<blocks2, 128, 0, stream>>>(
          W, types, Ew, Iw, spk, ncnt, st_ie, st_ii, out_ie, out_ii, N, T, t, B);
    } else {
      lif_update_spike<0><<<blocks1, 256, 0, stream>>>(
          delta_t, types, st_v, st_ie, st_ii, spk, out_s, out_v, N, T, t, BN);
      lif_matmul<0><<<blocks2, 128, 0, stream>>>(
          W, types, Ew, Iw, spk, ncnt, st_ie, st_ii, out_ie, out_ii, N, T, t, B);
    }
  }
}